// get_loss_74036646249099
// MI455X (gfx1250) — compile-verified
//
#include <hip/hip_runtime.h>
#include <hip/hip_bf16.h>
#include <math.h>

// CDNA5 WMMA vector types (wave32)
typedef __attribute__((ext_vector_type(2))) float v2f;
typedef __attribute__((ext_vector_type(8))) float v8f;

#define BS      8
#define HW      1024
#define NPTS    512
#define NSAMP   51      // hw // STEP = 1024 // 20
#define STEP    20
#define SCORW   0.01f

// ---------------------------------------------------------------------------
// Zero the per-(b,n) accumulators (graph-safe: no memset APIs).
// ---------------------------------------------------------------------------
__global__ void zero_ws_kernel(float* __restrict__ ws, int n) {
    int i = blockIdx.x * blockDim.x + threadIdx.x;
    if (i < n) ws[i] = 0.0f;
}

// ---------------------------------------------------------------------------
// One block per (batch, sample). 8 waves/block, each wave owns 4 M-tiles of
// 16 pred points. Squared distances via V_WMMA_F32_16X16X4_F32 with the
// row-constant |p|^2 hoisted OUT of the WMMA:
//   A[m][:] = [-2px, -2py, -2pz, 1 ]    (K = 4, fp32, exact)
//   B[:][n] = [ gx ,  gy ,  gz , gn]    gn = |g_n|^2
//   C       = 0                          (hoisted zero vector, no splat movs)
//   D       = gn - 2 p.g
//   min_q d^2 = |p|^2 + min_q D          (|p|^2 added once per M-tile)
//
// LDS B staging (layout-resolved, no lane-divergent selects):
//   ldsB[2q+0] = (gx, gy)   lanes 0..15  (K=0,1 slots)
//   ldsB[2q+1] = (gz, gn)   lanes 16..31 (K=2,3 slots)
// Inner loop = 1 ds_load_b64 + 1 v_wmma + 8 fmin (min3-fused).
// ---------------------------------------------------------------------------
__global__ __launch_bounds__(256) void chamfer_wmma_kernel(
    const float* __restrict__ pred_r,    // (BS,4,HW)
    const float* __restrict__ pred_t,    // (BS,3,HW)
    const float* __restrict__ gt_r,      // (BS,3,3)
    const float* __restrict__ gt_t,      // (BS,3)
    const float* __restrict__ model_xyz, // (BS,3,NPTS)
    float* __restrict__ ws_add)          // (BS*NSAMP) accumulators
{
    __shared__ float2 ldsB[NPTS * 2];    // 8 KB

    const int b    = blockIdx.x / NSAMP;
    const int n    = blockIdx.x % NSAMP;
    const int tid  = threadIdx.x;
    const int lane = tid & 31;
    const int wave = tid >> 5;
    const int lo16 = lane & 15;
    const int half = lane >> 4;          // 0: K=0,1 slots   1: K=2,3 slots

    const float* M = model_xyz + (size_t)b * 3 * NPTS;

    // ---- Stage GT points in WMMA B-operand order (cooperative) ----
    {
        const float* G = gt_r + b * 9;
        const float* T = gt_t + b * 3;
        for (int e = tid; e < NPTS * 2; e += 256) {
            int q = e >> 1;
            float mx = M[q], my = M[NPTS + q], mz = M[2 * NPTS + q];
            float gx = fmaf(G[0], mx, fmaf(G[1], my, fmaf(G[2], mz, T[0])));
            float gy = fmaf(G[3], mx, fmaf(G[4], my, fmaf(G[5], mz, T[1])));
            float gz = fmaf(G[6], mx, fmaf(G[7], my, fmaf(G[8], mz, T[2])));
            float gn = fmaf(gx, gx, fmaf(gy, gy, gz * gz));
            ldsB[e] = (e & 1) ? make_float2(gz, gn)
                              : make_float2(gx, gy);
        }
    }
    __syncthreads();

    // ---- Per-(b,n) pose: normalized quaternion -> rotation, translation ----
    const int idx = n * STEP;  // flattened h*w index of this sample
    float q0 = pred_r[(size_t)b * 4 * HW + 0 * HW + idx];
    float q1 = pred_r[(size_t)b * 4 * HW + 1 * HW + idx];
    float q2 = pred_r[(size_t)b * 4 * HW + 2 * HW + idx];
    float q3 = pred_r[(size_t)b * 4 * HW + 3 * HW + idx];
    float inv = 1.0f / sqrtf(fmaf(q0, q0, fmaf(q1, q1, fmaf(q2, q2, q3 * q3))));
    q0 *= inv; q1 *= inv; q2 *= inv; q3 *= inv;

    const float R00 = 1.0f - 2.0f * (q2 * q2 + q3 * q3);
    const float R01 = 2.0f * q1 * q2 - 2.0f * q0 * q3;
    const float R02 = 2.0f * q0 * q2 + 2.0f * q1 * q3;
    const float R10 = 2.0f * q1 * q2 + 2.0f * q3 * q0;
    const float R11 = 1.0f - 2.0f * (q1 * q1 + q3 * q3);
    const float R12 = -2.0f * q0 * q1 + 2.0f * q2 * q3;
    const float R20 = -2.0f * q0 * q2 + 2.0f * q1 * q3;
    const float R21 = 2.0f * q0 * q1 + 2.0f * q2 * q3;
    const float R22 = 1.0f - 2.0f * (q1 * q1 + q2 * q2);

    const float t0 = pred_t[(size_t)b * 3 * HW + 0 * HW + idx];
    const float t1 = pred_t[(size_t)b * 3 * HW + 1 * HW + idx];
    const float t2 = pred_t[(size_t)b * 3 * HW + 2 * HW + idx];

    // Per-lane base into the staged B operand (uniform inner-loop addressing)
    const float2* Bbase = &ldsB[(lo16 << 1) | half];

    const float BIG = 3.0e38f;
    const v8f Czero = { 0.0f, 0.0f, 0.0f, 0.0f, 0.0f, 0.0f, 0.0f, 0.0f };
    float waveSum = 0.0f;

    // ---- Each wave: 4 M-tiles of 16 pred points ----
    for (int i = 0; i < 4; ++i) {
        const int mtile = wave * 4 + i;
        const int p = mtile * 16 + lo16;

        float mx = M[p], my = M[NPTS + p], mz = M[2 * NPTS + p];
        float px = fmaf(R00, mx, fmaf(R01, my, fmaf(R02, mz, t0)));
        float py = fmaf(R10, mx, fmaf(R11, my, fmaf(R12, mz, t1)));
        float pz = fmaf(R20, mx, fmaf(R21, my, fmaf(R22, mz, t2)));
        float pn = fmaf(px, px, fmaf(py, py, pz * pz));  // |p|^2, row const

        // A-matrix 16x4 fp32 layout: lanes 0-15 hold K=0,1; lanes 16-31 K=2,3
        v2f A;
        A.x = half ? (-2.0f * pz) : (-2.0f * px);
        A.y = half ? 1.0f         : (-2.0f * py);

        v8f runmin = { BIG, BIG, BIG, BIG, BIG, BIG, BIG, BIG };

        #pragma unroll 8
        for (int nt = 0; nt < 32; ++nt) {
            // One uniform ds_load_b64; (g.x, g.y) is an even-aligned VGPR
            // pair = the WMMA B operand for this half-wave.
            float2 g = Bbase[nt * 32];

            v2f Bv;
            Bv.x = g.x;
            Bv.y = g.y;

            v8f D = __builtin_amdgcn_wmma_f32_16x16x4_f32(
                /*neg_a=*/false, A, /*neg_b=*/false, Bv,
                /*c_mod=*/(short)0, Czero, /*reuse_a=*/false, /*reuse_b=*/false);

            #pragma unroll
            for (int r = 0; r < 8; ++r)
                runmin[r] = fminf(runmin[r], D[r]);
        }

        // Min over the 16 columns covered by each 16-lane half -> full 512-col
        // min for rows r + 8*half
        #pragma unroll
        for (int r = 0; r < 8; ++r) {
            float v = runmin[r];
            v = fminf(v, __shfl_xor(v, 1, 32));
            v = fminf(v, __shfl_xor(v, 2, 32));
            v = fminf(v, __shfl_xor(v, 4, 32));
            v = fminf(v, __shfl_xor(v, 8, 32));
            runmin[r] = v;
        }

        // Add back the row-constant |p|^2 (pn of row r+8*half lives in lane
        // r+8*half; fetch with a bpermute-backed shuffle), sqrt, accumulate.
        float s8 = 0.0f;
        #pragma unroll
        for (int r = 0; r < 8; ++r) {
            float pnr = __shfl(pn, r + 8 * half, 32);
            s8 += sqrtf(fmaxf(runmin[r] + pnr, 0.0f));
        }
        s8 += __shfl_xor(s8, 16, 32);   // full 16-row sum in every lane
        waveSum += s8;
    }

    if (lane == 0)
        atomicAdd(&ws_add[blockIdx.x], waveSum);
}

// ---------------------------------------------------------------------------
// Finalize: add_ij = sum/512; loss = mean(add_ij*ps - 0.01*log(ps)); nan/inf->0
// ---------------------------------------------------------------------------
__global__ __launch_bounds__(512) void finalize_kernel(
    const float* __restrict__ ws_add,
    const float* __restrict__ pred_s,   // (BS,1,HW)
    float* __restrict__ out)
{
    __shared__ float red[512];
    const int tid = threadIdx.x;
    float v = 0.0f;
    if (tid < BS * NSAMP) {
        int b = tid / NSAMP;
        int n = tid % NSAMP;
        float add_ij = ws_add[tid] * (1.0f / (float)NPTS);
        float ps = pred_s[(size_t)b * HW + n * STEP];
        v = add_ij * ps - SCORW * logf(ps);
    }
    red[tid] = v;
    __syncthreads();
    for (int s = 256; s > 0; s >>= 1) {
        if (tid < s) red[tid] += red[tid + s];
        __syncthreads();
    }
    if (tid == 0) {
        float loss = red[0] / (float)(BS * NSAMP);
        if (isnan(loss) || isinf(loss)) loss = 0.0f;
        out[0] = loss;
    }
}

// ---------------------------------------------------------------------------
extern "C" void kernel_launch(void* const* d_in, const int* in_sizes, int n_in,
                              void* d_out, int out_size, void* d_ws, size_t ws_size,
                              hipStream_t stream) {
    (void)in_sizes; (void)n_in; (void)out_size; (void)ws_size;

    const float* pred_r    = (const float*)d_in[0];
    const float* pred_t    = (const float*)d_in[1];
    const float* pred_s    = (const float*)d_in[2];
    // d_in[3] = mask (unused by the reference math)
    const float* gt_r      = (const float*)d_in[4];
    const float* gt_t      = (const float*)d_in[5];
    const float* model_xyz = (const float*)d_in[6];
    // d_in[7] = cls_ids (unused)

    float* ws  = (float*)d_ws;   // BS*NSAMP accumulators
    float* out = (float*)d_out;

    zero_ws_kernel<<<1, 512, 0, stream>>>(ws, BS * NSAMP);
    chamfer_wmma_kernel<<<BS * NSAMP, 256, 0, stream>>>(
        pred_r, pred_t, gt_r, gt_t, model_xyz, ws);
    finalize_kernel<<<1, 512, 0, stream>>>(ws, pred_s, out);
}